// WireframeDetector_88364657147919
// MI455X (gfx1250) — compile-verified
//
#include <hip/hip_runtime.h>
#include <hip/hip_bf16.h>
#include <math.h>

// ---------------------------------------------------------------------------
// WireframeDetector decode: 3x3 NMS + stable descending sort + offset decode.
//   B=64 images, H=W=512, N=262144 pixels/image.
//   Pipeline:
//     1) nms_key_kernel    : async-LDS-tiled 3x3 max stencil -> packed
//                            (sortKey<<32 | idx) u64 records (workspace bufA)
//     2) 4x 8-bit LSD radix passes on the key word (hist -> scan -> stable
//                            scatter with a 64KB LDS per-thread rank table)
//     3) decode_kernel     : un-transform key -> score, gather offsets,
//                            write [B, N, 3] (y+yo+0.5, x+xo+0.5, score)
//   Workspace: 128MB bufA + 128MB bufB + 16MB histograms = 272MB.
// ---------------------------------------------------------------------------

#define B_IMG 64
#define H_DIM 512
#define W_DIM 512
#define N_PIX (H_DIM * W_DIM)

#define CHUNK 1024
#define CHUNKS (N_PIX / CHUNK) /* 256 chunk-blocks per image */
#define SCAT_THREADS 128
#define SCAT_IPT 8 /* 128 * 8 = 1024 = CHUNK */

// --- CDNA5 async global->LDS path (guarded: falls back to plain loads) -----
#if defined(__has_builtin)
#if __has_builtin(__builtin_amdgcn_global_load_async_to_lds_b32) && \
    __has_builtin(__builtin_amdgcn_s_wait_asynccnt)
#define USE_ASYNC_LDS 1
#endif
#endif
#ifndef USE_ASYNC_LDS
#define USE_ASYNC_LDS 0
#endif

#if USE_ASYNC_LDS
// Builtin signature (per hipcc diagnostic): params are int* in AS1 / AS3.
typedef __attribute__((address_space(1))) int gint_t;
typedef __attribute__((address_space(3))) int lint_t;
// Generic->AS casts via integer round-trip (C++ forbids direct AS casts).
// Generic LDS pointers carry the LDS byte offset in their low 32 bits.
__device__ __forceinline__ gint_t* to_gas(const void* p) {
  return (gint_t*)(unsigned long long)p;
}
__device__ __forceinline__ lint_t* to_las(void* p) {
  return (lint_t*)(unsigned int)(unsigned long long)p;
}
#endif

// ---------------------------------------------------------------------------
// 1) NMS + sort-key build.  32x32 output tile, 34x34 halo staged in LDS via
//    GLOBAL_LOAD_ASYNC_TO_LDS_B32 (ASYNCcnt) when available.
//    Key transform: k' = ~orderedBits(f) so ascending u32 == descending float.
// ---------------------------------------------------------------------------
__global__ __launch_bounds__(256) void nms_key_kernel(
    const float* __restrict__ hm, unsigned long long* __restrict__ recs) {
  __shared__ float tile[34 * 34];
  const int img = blockIdx.z;
  const int x0 = blockIdx.x * 32;
  const int y0 = blockIdx.y * 32;
  const int t = threadIdx.x;
  const float* img_hm = hm + (size_t)img * N_PIX;

  for (int h = t; h < 34 * 34; h += 256) {
    const int hy = h / 34;
    const int hx = h - hy * 34;
    const int gy = y0 + hy - 1;
    const int gx = x0 + hx - 1;
    if (gy >= 0 && gy < H_DIM && gx >= 0 && gx < W_DIM) {
#if USE_ASYNC_LDS
      __builtin_amdgcn_global_load_async_to_lds_b32(
          to_gas(img_hm + (size_t)gy * W_DIM + gx), to_las(&tile[h]), 0, 0);
#else
      tile[h] = img_hm[(size_t)gy * W_DIM + gx];
#endif
    } else {
      tile[h] = -INFINITY;  // jnp.pad(..., -inf): border never suppresses
    }
  }
#if USE_ASYNC_LDS
  __builtin_amdgcn_s_wait_asynccnt(0);
#endif
  __syncthreads();

  const int tx = t & 31;
  const int ty = t >> 5;  // 0..7, each thread does 4 rows
#pragma unroll
  for (int k = 0; k < 4; ++k) {
    const int ly = ty + k * 8;
    const float* r0 = &tile[ly * 34 + tx];
    const float* r1 = r0 + 34;
    const float* r2 = r1 + 34;
    const float c = r1[1];
    const float m = fmaxf(fmaxf(fmaxf(r0[0], r0[1]), fmaxf(r0[2], r1[0])),
                          fmaxf(fmaxf(r1[2], r2[0]), fmaxf(r2[1], r2[2])));
    const float jm = (m >= c) ? 0.6f * c : c;  // original-value NMS
    const unsigned u = __float_as_uint(jm);
    const unsigned s = (u & 0x80000000u) ? ~u : (u | 0x80000000u);
    const unsigned kp = ~s;  // ascending kp == descending float
    const unsigned idx = (unsigned)((y0 + ly) * W_DIM + (x0 + tx));
    recs[(size_t)img * N_PIX + idx] = ((unsigned long long)kp << 32) | idx;
  }
}

// ---------------------------------------------------------------------------
// 2a) Per-chunk 256-bin histogram (LDS atomics -> ds_add).
// ---------------------------------------------------------------------------
__global__ __launch_bounds__(256) void radix_hist_kernel(
    const unsigned long long* __restrict__ src, unsigned int* __restrict__ hist,
    int shift) {
  __shared__ unsigned int h[256];
  const int t = threadIdx.x;
  const int blk = blockIdx.x;
  const int img = blockIdx.y;
  h[t] = 0;
  __syncthreads();
  const unsigned long long* p = src + (size_t)img * N_PIX + (size_t)blk * CHUNK;
#pragma unroll
  for (int j = 0; j < CHUNK / 256; ++j) {
    const unsigned d = (unsigned)(p[t + j * 256] >> shift) & 255u;
    atomicAdd(&h[d], 1u);
  }
  __syncthreads();
  hist[((size_t)img * CHUNKS + blk) * 256 + t] = h[t];
}

// ---------------------------------------------------------------------------
// 2b) Per-image scan: hist[img][blk][d] -> global base for (blk, d).
//     Thread d owns digit column d; one block per image.
// ---------------------------------------------------------------------------
__global__ __launch_bounds__(256) void radix_scan_kernel(
    unsigned int* __restrict__ hist) {
  __shared__ unsigned int s[257];
  const int d = threadIdx.x;
  const int img = blockIdx.x;
  unsigned int* hb = hist + (size_t)img * CHUNKS * 256;
  unsigned tot = 0;
  for (int b = 0; b < CHUNKS; ++b) tot += hb[(size_t)b * 256 + d];
  s[d + 1] = tot;
  if (d == 0) s[0] = 0;
  __syncthreads();
  if (d == 0) {
    for (int i = 1; i <= 256; ++i) s[i] += s[i - 1];  // exclusive digit bases
  }
  __syncthreads();
  unsigned run = s[d];
  for (int b = 0; b < CHUNKS; ++b) {
    const unsigned c = hb[(size_t)b * 256 + d];
    hb[(size_t)b * 256 + d] = run;  // overwrite with global block offset
    run += c;
  }
}

// ---------------------------------------------------------------------------
// 2c) Stable scatter.  Each thread owns a contiguous run of 8 records;
//     a [128][256] u16 LDS rank table (64KB, fine in CDNA5's 320KB WGP LDS)
//     gives exact stable ranks: block order -> thread order -> item order.
// ---------------------------------------------------------------------------
__global__ __launch_bounds__(SCAT_THREADS) void radix_scatter_kernel(
    const unsigned long long* __restrict__ src,
    unsigned long long* __restrict__ dst, const unsigned int* __restrict__ hist,
    int shift) {
  extern __shared__ unsigned short cnt[];  // [SCAT_THREADS][256] u16
  __shared__ unsigned int base[256];
  const int t = threadIdx.x;
  const int blk = blockIdx.x;
  const int img = blockIdx.y;

  for (int d = 0; d < 256; ++d) cnt[t * 256 + d] = 0;  // own row, no race
  for (int d = t; d < 256; d += SCAT_THREADS)
    base[d] = hist[((size_t)img * CHUNKS + blk) * 256 + d];

  const unsigned long long* p =
      src + (size_t)img * N_PIX + (size_t)blk * CHUNK + (size_t)t * SCAT_IPT;
  unsigned long long rec[SCAT_IPT];
  unsigned dig[SCAT_IPT];
#pragma unroll
  for (int j = 0; j < SCAT_IPT; ++j) {
    rec[j] = p[j];
    dig[j] = (unsigned)(rec[j] >> shift) & 255u;
    cnt[t * 256 + dig[j]]++;  // own row
  }
  __syncthreads();

  // Exclusive prefix over threads (in thread order) per digit column.
  for (int d = t; d < 256; d += SCAT_THREADS) {
    unsigned run = 0;
    for (int tt = 0; tt < SCAT_THREADS; ++tt) {
      const unsigned c = cnt[tt * 256 + d];
      cnt[tt * 256 + d] = (unsigned short)run;
      run += c;
    }
  }
  __syncthreads();

  unsigned long long* dimg = dst + (size_t)img * N_PIX;
#pragma unroll
  for (int j = 0; j < SCAT_IPT; ++j) {
    const unsigned d = dig[j];
    const unsigned short lr = cnt[t * 256 + d];
    cnt[t * 256 + d] = (unsigned short)(lr + 1u);  // own row
    dimg[base[d] + lr] = rec[j];
  }
}

// ---------------------------------------------------------------------------
// 3) Decode: rank r -> (y+yo+0.5, x+xo+0.5, score).
// ---------------------------------------------------------------------------
__global__ __launch_bounds__(256) void decode_kernel(
    const unsigned long long* __restrict__ sorted,
    const float* __restrict__ off, float* __restrict__ out) {
  const int img = blockIdx.y;
  const int r = blockIdx.x * 256 + threadIdx.x;
  const unsigned long long rec = sorted[(size_t)img * N_PIX + r];
  const unsigned kp = (unsigned)(rec >> 32);
  const unsigned s = ~kp;  // invert key transform
  const unsigned u = (s & 0x80000000u) ? (s ^ 0x80000000u) : ~s;
  const float score = __uint_as_float(u);
  const unsigned idx = (unsigned)rec;
  const unsigned y = idx >> 9;
  const unsigned x = idx & 511u;
  const float* oimg = off + (size_t)img * 2 * N_PIX;
  const float xo = oimg[idx];          // offset[img][0] = x-offset
  const float yo = oimg[N_PIX + idx];  // offset[img][1] = y-offset
  float* o = out + ((size_t)img * N_PIX + (size_t)r) * 3;
  o[0] = ((float)y + yo) + 0.5f;
  o[1] = ((float)x + xo) + 0.5f;
  o[2] = score;
}

// ---------------------------------------------------------------------------
extern "C" void kernel_launch(void* const* d_in, const int* in_sizes, int n_in,
                              void* d_out, int out_size, void* d_ws,
                              size_t ws_size, hipStream_t stream) {
  (void)in_sizes;
  (void)n_in;
  (void)out_size;
  (void)ws_size;
  const float* hm = (const float*)d_in[0];   // [64,512,512] f32
  const float* off = (const float*)d_in[1];  // [64,2,512,512] f32
  float* out = (float*)d_out;                // [64,262144,3] f32

  unsigned long long* bufA = (unsigned long long*)d_ws;          // 128 MB
  unsigned long long* bufB = bufA + (size_t)B_IMG * N_PIX;       // 128 MB
  unsigned int* hist = (unsigned int*)(bufB + (size_t)B_IMG * N_PIX);  // 16 MB

  nms_key_kernel<<<dim3(W_DIM / 32, H_DIM / 32, B_IMG), 256, 0, stream>>>(hm,
                                                                          bufA);

  unsigned long long* src = bufA;
  unsigned long long* dst = bufB;
  for (int pass = 0; pass < 4; ++pass) {
    const int shift = 32 + 8 * pass;  // key lives in bits [63:32]
    radix_hist_kernel<<<dim3(CHUNKS, B_IMG), 256, 0, stream>>>(src, hist,
                                                               shift);
    radix_scan_kernel<<<B_IMG, 256, 0, stream>>>(hist);
    radix_scatter_kernel<<<dim3(CHUNKS, B_IMG), SCAT_THREADS,
                           SCAT_THREADS * 256 * sizeof(unsigned short),
                           stream>>>(src, dst, hist, shift);
    unsigned long long* tmp = src;
    src = dst;
    dst = tmp;
  }
  // 4 passes: A->B->A->B->A, final sorted data back in bufA (== src).
  decode_kernel<<<dim3(N_PIX / 256, B_IMG), 256, 0, stream>>>(src, off, out);
}